// ProbabilisticGNN_85907935854681
// MI455X (gfx1250) — compile-verified
//
#include <hip/hip_runtime.h>
#include <math.h>

// ---------------------------------------------------------------------------
// 2-layer GCN (eval):  out = log_softmax( GCN2( relu(GCN1(x)) ) )
// N = 100000 nodes, E = 6.4M edges. Memory/atomic bound; edge_index (102.4MB)
// fits MI455X's 192MB L2, so the three edge passes stream from L2 after the
// first touch. The single dense GEMM [N,16]@[16,2] uses V_WMMA_F32_16X16X4_F32
// (exact fp32, matches reference numerics).
// ---------------------------------------------------------------------------

typedef float v2f __attribute__((ext_vector_type(2)));
typedef float v8f __attribute__((ext_vector_type(8)));

__device__ __forceinline__ void atomAddF(float* p, float v) {
    // hardware global_atomic_add_f32 (no CAS loop)
    unsafeAtomicAdd(p, v);
}

// ---- pass 0: zero the degree counters ------------------------------------
__global__ void k_zero_u32(unsigned* __restrict__ p, int n) {
    int i = blockIdx.x * blockDim.x + threadIdx.x;
    if (i < n) p[i] = 0u;
}

// ---- pass 1: in-degree count ---------------------------------------------
__global__ void k_count(const long long* __restrict__ dst,
                        unsigned* __restrict__ cnt, int E) {
    int e = blockIdx.x * blockDim.x + threadIdx.x;
    if (e < E) atomicAdd(&cnt[(int)dst[e]], 1u);
}

// ---- pass 2: per-node prep: dinv, xw1 = x*W1 (rank-1), agg1 = xw1/deg ----
__global__ void k_node_prep(const float* __restrict__ x,
                            const float* __restrict__ W1,
                            const unsigned* __restrict__ cnt,
                            float* __restrict__ dinv,
                            float* __restrict__ xw1,
                            float* __restrict__ agg1, int N) {
    int i = blockIdx.x * blockDim.x + threadIdx.x;
    if (i >= N) return;
    float degf = (float)cnt[i] + 1.0f;          // A + I in-degree
    dinv[i]    = rsqrtf(degf);
    float invd = 1.0f / degf;
    float xv   = x[i];
    float4* xw = (float4*)(xw1 + (size_t)i * 16);
    float4* ag = (float4*)(agg1 + (size_t)i * 16);
#pragma unroll
    for (int j = 0; j < 4; ++j) {
        float4 w = ((const float4*)W1)[j];
        float4 v = make_float4(xv * w.x, xv * w.y, xv * w.z, xv * w.w);
        xw[j] = v;
        ag[j] = make_float4(v.x * invd, v.y * invd, v.z * invd, v.w * invd);
    }
}

// ---- pass 3: edge scatter, layer 1 (16 features) -------------------------
__global__ void k_scatter1(const long long* __restrict__ src,
                           const long long* __restrict__ dst,
                           const float* __restrict__ dinv,
                           const float* __restrict__ xw1,
                           float* __restrict__ agg1, int E) {
    int e = blockIdx.x * blockDim.x + threadIdx.x;
    if (e >= E) return;
    int s = (int)src[e], d = (int)dst[e];
    float nrm = dinv[s] * dinv[d];
    const float4* xs = (const float4*)(xw1 + (size_t)s * 16);
    float* ad = agg1 + (size_t)d * 16;
#pragma unroll
    for (int j = 0; j < 4; ++j) {
        float4 v = xs[j];
        atomAddF(ad + 4 * j + 0, v.x * nrm);
        atomAddF(ad + 4 * j + 1, v.y * nrm);
        atomAddF(ad + 4 * j + 2, v.z * nrm);
        atomAddF(ad + 4 * j + 3, v.w * nrm);
    }
}

// ---- pass 4: fused  h=relu(agg1+b1);  xw2=h@W2 (WMMA f32 16x16x4);
//              agg2 = xw2/deg  (self-loop init) -----------------------------
// One wave32 computes a 16-node tile. A: 16x4 f32 per K-step (lane M=0..15,
// khalf=lane>>4 selects K pair). B: W2 [16,2] zero-padded to 4x16 per step,
// built with unconditional in-range loads + branchless cndmask (no exec
// divergence before the WMMAs).
__global__ __launch_bounds__(256) void k_l2_wmma(
    const float* __restrict__ agg1, const float* __restrict__ b1,
    const float* __restrict__ W2, const unsigned* __restrict__ cnt,
    float* __restrict__ xw2, float* __restrict__ agg2, int n_tiles, int N) {
    int wave = blockIdx.x * (blockDim.x >> 5) + (threadIdx.x >> 5);
    if (wave >= n_tiles) return;             // wave-uniform: EXEC stays all-1s
    int lane  = threadIdx.x & 31;
    int M     = lane & 15;
    int khalf = lane >> 4;
    int nb    = wave * 16;
    int row   = nb + M;
    if (row > N - 1) row = N - 1;            // tail clamp (N%16==0 here)
    const bool colActive = (M < 2);          // D columns 0..1 are real outputs
    const int  c         = M & 1;            // W2 column this lane would hold

    v8f acc = {};
#pragma unroll
    for (int k0 = 0; k0 < 16; k0 += 4) {
        int f0 = k0 + 2 * khalf;             // global feature of A/B .x
        // A-matrix element pair for this lane (relu(agg1+b1) fused here)
        float2 av = *(const float2*)(agg1 + (size_t)row * 16 + f0);
        float2 bv = *(const float2*)(b1 + f0);
        v2f A = {fmaxf(av.x + bv.x, 0.0f), fmaxf(av.y + bv.y, 0.0f)};
        // B-matrix: row K = f0 (.x) / f0+1 (.y), column n = lane&15.
        // Loads are unconditionally in-range; zero-pad via branchless select.
        float w0 = W2[f0 * 2 + c];
        float w1 = W2[(f0 + 1) * 2 + c];
        v2f B = {colActive ? w0 : 0.0f, colActive ? w1 : 0.0f};
        acc = __builtin_amdgcn_wmma_f32_16x16x4_f32(
            /*neg_a=*/false, A, /*neg_b=*/false, B,
            /*c_mod=*/(short)0, acc, /*reuse_a=*/false, /*reuse_b=*/false);
    }

    // D layout: VGPR g, lane L -> row M = g + 8*(L/16), col n = L&15.
    if (colActive) {
#pragma unroll
        for (int g = 0; g < 8; ++g) {
            int node = nb + g + 8 * khalf;
            if (node < N) {
                float v    = acc[g];
                float invd = 1.0f / ((float)cnt[node] + 1.0f);
                xw2[(size_t)node * 2 + c]  = v;
                agg2[(size_t)node * 2 + c] = v * invd;   // self-loop term
            }
        }
    }
}

// ---- pass 5: edge scatter, layer 2 (2 features) --------------------------
__global__ void k_scatter2(const long long* __restrict__ src,
                           const long long* __restrict__ dst,
                           const float* __restrict__ dinv,
                           const float* __restrict__ xw2,
                           float* __restrict__ agg2, int E) {
    int e = blockIdx.x * blockDim.x + threadIdx.x;
    if (e >= E) return;
    int s = (int)src[e], d = (int)dst[e];
    float nrm = dinv[s] * dinv[d];
    float2 v  = *(const float2*)(xw2 + (size_t)s * 2);
    atomAddF(agg2 + (size_t)d * 2 + 0, v.x * nrm);
    atomAddF(agg2 + (size_t)d * 2 + 1, v.y * nrm);
}

// ---- pass 6: +b2 and 2-class log-softmax ---------------------------------
__global__ void k_finalize(const float* __restrict__ agg2,
                           const float* __restrict__ b2,
                           float* __restrict__ out, int N) {
    int i = blockIdx.x * blockDim.x + threadIdx.x;
    if (i >= N) return;
    float z0 = agg2[(size_t)i * 2 + 0] + b2[0];
    float z1 = agg2[(size_t)i * 2 + 1] + b2[1];
    float m  = fmaxf(z0, z1);
    float l  = m + logf(expf(z0 - m) + expf(z1 - m));
    out[(size_t)i * 2 + 0] = z0 - l;
    out[(size_t)i * 2 + 1] = z1 - l;
}

extern "C" void kernel_launch(void* const* d_in, const int* in_sizes, int n_in,
                              void* d_out, int out_size, void* d_ws, size_t ws_size,
                              hipStream_t stream) {
    const float*     x   = (const float*)d_in[0];
    const long long* ei  = (const long long*)d_in[1];   // int64 [2,E]
    const float*     W1  = (const float*)d_in[2];
    const float*     b1  = (const float*)d_in[3];
    const float*     W2  = (const float*)d_in[4];
    const float*     b2  = (const float*)d_in[5];
    float*           out = (float*)d_out;

    const int N = in_sizes[0];
    const int E = in_sizes[1] / 2;
    const long long* src = ei;
    const long long* dst = ei + E;

    // workspace carve-out (256B aligned)
    char*  ws  = (char*)d_ws;
    size_t off = 0;
    auto carve = [&](size_t bytes) {
        void* p = ws + off;
        off = (off + bytes + 255) & ~(size_t)255;
        return p;
    };
    unsigned* cnt  = (unsigned*)carve((size_t)N * 4);
    float*    dinv = (float*)   carve((size_t)N * 4);
    float*    xw1  = (float*)   carve((size_t)N * 16 * 4);
    float*    agg1 = (float*)   carve((size_t)N * 16 * 4);
    float*    xw2  = (float*)   carve((size_t)N * 2 * 4);
    float*    agg2 = (float*)   carve((size_t)N * 2 * 4);

    const int T  = 256;
    const int gN = (N + T - 1) / T;
    const int gE = (E + T - 1) / T;

    k_zero_u32 <<<gN, T, 0, stream>>>(cnt, N);
    k_count    <<<gE, T, 0, stream>>>(dst, cnt, E);
    k_node_prep<<<gN, T, 0, stream>>>(x, W1, cnt, dinv, xw1, agg1, N);
    k_scatter1 <<<gE, T, 0, stream>>>(src, dst, dinv, xw1, agg1, E);

    const int n_tiles = (N + 15) / 16;                  // 6250 wave-tiles
    const int wavesPerBlock = T / 32;
    const int gW = (n_tiles + wavesPerBlock - 1) / wavesPerBlock;
    k_l2_wmma  <<<gW, T, 0, stream>>>(agg1, b1, W2, cnt, xw2, agg2, n_tiles, N);

    k_scatter2 <<<gE, T, 0, stream>>>(src, dst, dinv, xw2, agg2, E);
    k_finalize <<<gN, T, 0, stream>>>(agg2, b2, out, N);
}